// DWT_ms_69492570849964
// MI455X (gfx1250) — compile-verified
//
#include <hip/hip_runtime.h>

// 2-D Haar DWT (level 1) on (16, 4, 1024, 1024) fp32.
// D = H(16x4) * X(4x16) via V_WMMA_F32_16X16X4_F32: K=4 matches the Haar
// butterfly exactly. B operand is built by a single coalesced global_load_b64
// (lanes 0-15 read (a,b) from row 2y, lanes 16-31 read (c,d) from row 2y+1).
// Memory-bound: 512 MB streamed once -> NT hints, ~22 us floor @ 23.3 TB/s.
// UNROLL=8 amortizes the per-wave address setup over 4 KB of traffic.

typedef __attribute__((ext_vector_type(2))) float v2f;
typedef __attribute__((ext_vector_type(8))) float v8f;

#define WAVES_PER_BLOCK 8
#define UNROLL 8   // 8 groups of 16 output pixels per wave = 128 px along a row

__global__ __launch_bounds__(256) void haar_dwt_wmma_kernel(
    const float* __restrict__ in, float* __restrict__ out) {
    const int lane      = threadIdx.x & 31;   // wave32
    const int waveInBlk = threadIdx.x >> 5;
    const unsigned w    = blockIdx.x * WAVES_PER_BLOCK + waveInBlk;

    const int n    = lane & 15;  // D column / A row (M)
    const int half = lane >> 4;  // 0: holds K={0,1} (a,b) ; 1: K={2,3} (c,d)

    // wave -> (plane, output row, 128-px tile). 4 tiles cover a 512-px row.
    const unsigned tile  = w & 3;
    const unsigned row   = (w >> 2) & 511;
    const unsigned plane = w >> 11;          // 0..63 = B*C planes

    // ---- A matrix: 16x4 Haar analysis matrix, entries +-0.5 ----
    // rows (mod 4): LL:++++  LH:++--  HL:+-+-  HH:+--+   (cols a,b,c,d)
    const int   sub = n & 3;
    const float sb  = (sub & 2) ? -0.5f : 0.5f;
    const float sc  = (sub & 1) ? -0.5f : 0.5f;
    const float sd  = ((sub ^ (sub >> 1)) & 1) ? -0.5f : 0.5f;
    v2f amat;
    amat.x = half ? sc : 0.5f;  // K = 0 (a) or K = 2 (c)
    amat.y = half ? sd : sb;    // K = 1 (b) or K = 3 (d)

    // ---- B operand: one b64 load per group, 128B contiguous per half-wave ----
    const unsigned iy  = 2u * row + (unsigned)half;
    const float*   pin = in + ((size_t)plane << 20) + (size_t)iy * 1024u
                            + (size_t)tile * (UNROLL * 32) + 2u * (unsigned)n;
    v2f bm[UNROLL];
#pragma unroll
    for (int u = 0; u < UNROLL; ++u)
        bm[u] = __builtin_nontemporal_load((const v2f*)(pin + u * 32));

    // ---- output pointers: LL plane, then high=(16,12,512,512)=[LH|HL|HH] ----
    const size_t PLANE = 512u * 512u;
    const size_t LL_SZ = 64u * PLANE;
    const size_t bidx  = plane >> 2, ch = plane & 3;
    const size_t roff  = (size_t)row * 512u + (size_t)tile * (UNROLL * 16) + (size_t)n;
    float* pLL = out + (size_t)plane * PLANE + roff;
    float* pLH = out + LL_SZ + (bidx * 12 + 0 + ch) * PLANE + roff;
    float* pHL = out + LL_SZ + (bidx * 12 + 4 + ch) * PLANE + roff;
    float* pHH = out + LL_SZ + (bidx * 12 + 8 + ch) * PLANE + roff;
    // D duplicates all 4 subbands into both lane halves -> split stores:
    float* o0 = half ? pLH : pLL;  // lanes 0-15: LL, lanes 16-31: LH
    float* o1 = half ? pHH : pHL;  // lanes 0-15: HL, lanes 16-31: HH

#pragma unroll
    for (int u = 0; u < UNROLL; ++u) {
        v8f acc = {};
        // (neg_a, A, neg_b, B, c_mod, C, reuse_a, reuse_b)
        acc = __builtin_amdgcn_wmma_f32_16x16x4_f32(
            false, amat, false, bm[u], (short)0, acc, false, false);
        const float v0 = half ? acc[1] : acc[0];  // LL / LH
        const float v1 = half ? acc[3] : acc[2];  // HL / HH
        __builtin_nontemporal_store(v0, o0 + u * 16);
        __builtin_nontemporal_store(v1, o1 + u * 16);
    }
}

extern "C" void kernel_launch(void* const* d_in, const int* in_sizes, int n_in,
                              void* d_out, int out_size, void* d_ws, size_t ws_size,
                              hipStream_t stream) {
    const float* in = (const float*)d_in[0];
    float* out = (float*)d_out;
    // waves total = 64 planes * 512 rows * 4 tiles = 131,072 ; 8 waves/block
    dim3 grid(131072 / WAVES_PER_BLOCK), block(WAVES_PER_BLOCK * 32);
    hipLaunchKernelGGL(haar_dwt_wmma_kernel, grid, block, 0, stream, in, out);
}